// Criterion_36945308680563
// MI455X (gfx1250) — compile-verified
//
#include <hip/hip_runtime.h>

typedef float v2f __attribute__((ext_vector_type(2)));
typedef float v4f __attribute__((ext_vector_type(4)));
typedef float v8f __attribute__((ext_vector_type(8)));

#define NB     2
#define NPTS   16384
#define NOBS   6000
#define NFACES 8192
#define PT_TILES   (NPTS / 16)    // 1024
#define FACE_TILES (NFACES / 16)  // 512

#define KEY_MASK 0xFFFFE000u      // keep f32 sign+exp+10 mantissa bits
#define IDX_MASK 0x00001FFFu      // 13 bits: face index 0..8191

__device__ __forceinline__ unsigned umin32(unsigned a, unsigned b) {
    return a < b ? a : b;
}

// ---------------------------------------------------------------------------
// Kernel 1: face centers, normals, and interleaved WMMA A-operands.
// Per face f: faceOp[f][8] = { P0,P1, C0,C1, P2,P3, C2,C3 } where
// P = (-2m, |m|^2) for prev center, C likewise for curr center.
// K-half 0 reads bytes [0,16), K-half 1 reads bytes [16,32): one b128 each.
// ---------------------------------------------------------------------------
__global__ void face_prep_kernel(const float* __restrict__ o_prev,
                                 const float* __restrict__ o_curr,
                                 const int*   __restrict__ faces,
                                 float* __restrict__ faceOp,
                                 float* __restrict__ fpC,
                                 float* __restrict__ fcC,
                                 float* __restrict__ nP,
                                 float* __restrict__ nC)
{
    int g = blockIdx.x * blockDim.x + threadIdx.x;
    if (g >= NB * NFACES) return;
    int b = g / NFACES;
    const int* fi = faces + (size_t)g * 3;
    int i0 = fi[0], i1 = fi[1], i2 = fi[2];

    const float* bases[2] = { o_prev + (size_t)b * NOBS * 3,
                              o_curr + (size_t)b * NOBS * 3 };
    float* ctr[2] = { fpC, fcC };
    float* nrm[2] = { nP,  nC  };
    float op[2][4];

    #pragma unroll
    for (int s = 0; s < 2; ++s) {
        const float* o = bases[s];
        float ax = o[i0*3+0], ay = o[i0*3+1], az = o[i0*3+2];
        float bx = o[i1*3+0], by = o[i1*3+1], bz = o[i1*3+2];
        float cx = o[i2*3+0], cy = o[i2*3+1], cz = o[i2*3+2];
        float mx = (ax + bx + cx) * (1.0f/3.0f);
        float my = (ay + by + cy) * (1.0f/3.0f);
        float mz = (az + bz + cz) * (1.0f/3.0f);
        float e1x = bx-ax, e1y = by-ay, e1z = bz-az;
        float e2x = cx-ax, e2y = cy-ay, e2z = cz-az;
        float nx = e1y*e2z - e1z*e2y;
        float ny = e1z*e2x - e1x*e2z;
        float nz = e1x*e2y - e1y*e2x;
        float inv = 1.0f / fmaxf(sqrtf(nx*nx + ny*ny + nz*nz), 1e-12f);
        nx *= inv; ny *= inv; nz *= inv;

        op[s][0] = -2.0f * mx;
        op[s][1] = -2.0f * my;
        op[s][2] = -2.0f * mz;
        op[s][3] = mx*mx + my*my + mz*mz;

        size_t o4 = (size_t)g * 4;
        ctr[s][o4+0] = mx; ctr[s][o4+1] = my; ctr[s][o4+2] = mz; ctr[s][o4+3] = 0.f;
        nrm[s][o4+0] = nx; nrm[s][o4+1] = ny; nrm[s][o4+2] = nz; nrm[s][o4+3] = 0.f;
    }

    float* fo = faceOp + (size_t)g * 8;
    fo[0] = op[0][0]; fo[1] = op[0][1]; fo[2] = op[1][0]; fo[3] = op[1][1];
    fo[4] = op[0][2]; fo[5] = op[0][3]; fo[6] = op[1][2]; fo[7] = op[1][3];
}

// ---------------------------------------------------------------------------
// Kernel 2: brute-force 1-NN via V_WMMA_F32_16X16X4_F32.
//   A (16x4) = faces (-2q, |q|^2); B (4x16) = points (x,y,z,1);
//   C        = per-lane broadcast of |p|^2  ->  D[m][n] = |p_n - q_m|^2 >= 0.
// Argmin = unsigned min over packed keys (d2 bits | 13-bit face index).
// ---------------------------------------------------------------------------
__global__ void __launch_bounds__(256)
knn_wmma_kernel(const float* __restrict__ prev_pos,
                const float* __restrict__ curr_pos,
                const float* __restrict__ faceOp,
                unsigned* __restrict__ keyP,
                unsigned* __restrict__ keyC)
{
    const int lane  = threadIdx.x & 31;
    const int gwave = blockIdx.x * (blockDim.x >> 5) + (threadIdx.x >> 5);
    const int b     = gwave / PT_TILES;
    const int ptile = gwave % PT_TILES;
    const int col   = lane & 15;
    const bool lowk = (lane < 16);

    // --- B operand (points, padded to (x,y,z,1)) + |p|^2 for the C operand ---
    const int pidx = ptile * 16 + col;
    const float* pp = prev_pos + ((size_t)b * NPTS + pidx) * 3;
    const float* pc = curr_pos + ((size_t)b * NPTS + pidx) * 3;
    float ppx = pp[0], ppy = pp[1], ppz = pp[2];
    float pcx = pc[0], pcy = pc[1], pcz = pc[2];

    v2f bP, bC;
    bP[0] = lowk ? ppx : ppz;  bP[1] = lowk ? ppy : 1.0f;
    bC[0] = lowk ? pcx : pcz;  bC[1] = lowk ? pcy : 1.0f;

    float pnP = ppx*ppx + ppy*ppy + ppz*ppz;
    float pnC = pcx*pcx + pcy*pcy + pcz*pcz;
    v8f cP = {pnP,pnP,pnP,pnP,pnP,pnP,pnP,pnP};
    v8f cC = {pnC,pnC,pnC,pnC,pnC,pnC,pnC,pnC};

    // this lane's K-half slice of the interleaved face operands
    const float* fop = faceOp + (size_t)b * NFACES * 8 + (lowk ? 0 : 4);
    const unsigned half = lowk ? 0u : 8u;

    unsigned gP = 0xFFFFFFFFu;
    unsigned gC = 0xFFFFFFFFu;

    for (int t = 0; t < FACE_TILES; ++t) {
        // one b128: {aP0,aP1,aC0,aC1} for face row (t*16+col), this K-half
        v4f fo = *(const v4f*)(fop + (size_t)(t * 16 + col) * 8);
        v2f aP; aP[0] = fo[0]; aP[1] = fo[1];
        v2f aC; aC[0] = fo[2]; aC[1] = fo[3];

        v8f dP = __builtin_amdgcn_wmma_f32_16x16x4_f32(
            false, aP, false, bP, (short)0, cP, false, false);
        v8f dC = __builtin_amdgcn_wmma_f32_16x16x4_f32(
            false, aC, false, bC, (short)0, cC, false, false);

        const unsigned tb = (unsigned)(t * 16) + half;  // low 3 bits are 0
        #pragma unroll
        for (int v = 0; v < 8; v += 2) {
            unsigned i0 = tb | (unsigned)v;
            unsigned i1 = tb | (unsigned)(v + 1);
            unsigned kP0 = (__float_as_uint(dP[v])   & KEY_MASK) | i0;
            unsigned kP1 = (__float_as_uint(dP[v+1]) & KEY_MASK) | i1;
            gP = umin32(umin32(kP0, kP1), gP);          // -> v_min3_u32
            unsigned kC0 = (__float_as_uint(dC[v])   & KEY_MASK) | i0;
            unsigned kC1 = (__float_as_uint(dC[v+1]) & KEY_MASK) | i1;
            gC = umin32(umin32(kC0, kC1), gC);
        }
    }

    // merge row-halves: lane n holds M=0-7 candidates, lane n+16 holds M=8-15
    gP = umin32(gP, (unsigned)__shfl_xor((int)gP, 16, 32));
    gC = umin32(gC, (unsigned)__shfl_xor((int)gC, 16, 32));

    if (lowk) {
        size_t o = (size_t)b * NPTS + pidx;
        keyP[o] = gP;
        keyC[o] = gC;
    }
}

// ---------------------------------------------------------------------------
// Kernel 3: unpack keys, gather nearest-face data, friction, reduce.
// ---------------------------------------------------------------------------
__global__ void finalize_kernel(const float* __restrict__ prev_pos,
                                const float* __restrict__ curr_pos,
                                const float* __restrict__ v_mass,
                                const unsigned* __restrict__ keyP,
                                const unsigned* __restrict__ keyC,
                                const float* __restrict__ fpC, const float* __restrict__ fcC,
                                const float* __restrict__ nP,  const float* __restrict__ nC,
                                float* __restrict__ out)
{
    __shared__ float red[256];
    int g = blockIdx.x * blockDim.x + threadIdx.x;
    float acc = 0.0f;
    if (g < NB * NPTS) {
        int b = g / NPTS;
        const float* p = prev_pos + (size_t)g * 3;
        const float* c = curr_pos + (size_t)g * 3;
        float px = p[0], py = p[1], pz = p[2];
        float cx = c[0], cy = c[1], cz = c[2];

        unsigned kp = keyP[g], kc = keyC[g];
        float d2p = __uint_as_float(kp & KEY_MASK);
        float d2c = __uint_as_float(kc & KEY_MASK);
        int ip = (int)(kp & IDX_MASK);
        int ic = (int)(kc & IDX_MASK);

        size_t fb = (size_t)b * NFACES;
        const float* npp = fpC + (fb + ip) * 4;
        const float* npc = fcC + (fb + ip) * 4;
        const float* n1  = nP  + (fb + ip) * 4;
        const float* n2  = nC  + (fb + ic) * 4;

        float nmx = 0.5f * (n1[0] + n2[0]);
        float nmy = 0.5f * (n1[1] + n2[1]);
        float nmz = 0.5f * (n1[2] + n2[2]);

        bool mask = (d2p < 0.003f) && (d2c < 0.003f);

        float dx = cx - (px + (npc[0] - npp[0]));
        float dy = cy - (py + (npc[1] - npp[1]));
        float dz = cz - (pz + (npc[2] - npp[2]));
        float pr = dx*nmx + dy*nmy + dz*nmz;
        float dpx = dx - pr*nmx;
        float dpy = dy - pr*nmy;
        float dpz = dz - pr*nmz;
        float dnorm = sqrtf(dpx*dpx + dpy*dpy + dpz*dpz);
        float cosn  = fabsf(dpz) / fmaxf(dnorm, 1e-8f);
        float cosp  = sqrtf(fabsf(1.0f - cosn*cosn) + 1e-4f);
        acc = v_mass[g] * cosp * 9.81f * dnorm * (mask ? 1.0f : 0.0f);
    }
    red[threadIdx.x] = acc;
    __syncthreads();
    for (int s = blockDim.x >> 1; s > 0; s >>= 1) {
        if ((int)threadIdx.x < s) red[threadIdx.x] += red[threadIdx.x + s];
        __syncthreads();
    }
    if (threadIdx.x == 0) atomicAdd(out, red[0] * 0.5f);  // WEIGHT * MU / B
}

__global__ void zero_out_kernel(float* out) { out[0] = 0.0f; }

// ---------------------------------------------------------------------------
extern "C" void kernel_launch(void* const* d_in, const int* in_sizes, int n_in,
                              void* d_out, int out_size, void* d_ws, size_t ws_size,
                              hipStream_t stream)
{
    const float* prev_pos = (const float*)d_in[0];
    const float* curr_pos = (const float*)d_in[1];
    const float* v_mass   = (const float*)d_in[2];
    const float* o_prev   = (const float*)d_in[3];
    const float* o_curr   = (const float*)d_in[4];
    const int*   faces    = (const int*)  d_in[5];
    float* out = (float*)d_out;

    // Workspace layout
    float* ws = (float*)d_ws;
    const size_t F4 = (size_t)NB * NFACES * 4;
    const size_t F8 = (size_t)NB * NFACES * 8;
    const size_t P1 = (size_t)NB * NPTS;
    float*    faceOp = ws;            ws += F8;
    float*    fpC    = ws;            ws += F4;
    float*    fcC    = ws;            ws += F4;
    float*    nP     = ws;            ws += F4;
    float*    nC     = ws;            ws += F4;
    unsigned* keyP   = (unsigned*)ws; ws += P1;
    unsigned* keyC   = (unsigned*)ws;

    // 1) face centers / normals / interleaved WMMA operands
    face_prep_kernel<<<(NB * NFACES + 255) / 256, 256, 0, stream>>>(
        o_prev, o_curr, faces, faceOp, fpC, fcC, nP, nC);

    // 2) WMMA brute-force 1-NN: 2048 waves, 8 waves/block
    knn_wmma_kernel<<<(NB * PT_TILES) / 8, 256, 0, stream>>>(
        prev_pos, curr_pos, faceOp, keyP, keyC);

    // 3) finalize + reduce
    zero_out_kernel<<<1, 1, 0, stream>>>(out);
    finalize_kernel<<<(NB * NPTS + 255) / 256, 256, 0, stream>>>(
        prev_pos, curr_pos, v_mass, keyP, keyC, fpC, fcC, nP, nC, out);
}